// SCELoss_71313636982876
// MI455X (gfx1250) — compile-verified
//
#include <hip/hip_runtime.h>
#include <hip/hip_bf16.h>

#ifndef __has_builtin
#define __has_builtin(x) 0
#endif

#define SCE_C    128
#define SCE_NB   15
#define SCE_NSEG (SCE_C * SCE_NB)   // 1920
#define WAVES_PER_BLOCK 8
#define BLOCK_THREADS   256
#define SCE_DEPTH 8                 // async ring depth (7 loads in flight / wave)

typedef float v4f __attribute__((ext_vector_type(4)));

#if __has_builtin(__builtin_amdgcn_global_load_async_to_lds_b128) && \
    __has_builtin(__builtin_amdgcn_s_wait_asynccnt)
#define SCE_ASYNC 1
#else
#define SCE_ASYNC 0
#endif

#if SCE_ASYNC
// Builtin signature (from hipcc diagnostic): params are AS1 / AS3 pointers to
// 16-byte int vectors, plus imm offset and imm cpol.
typedef int v4i_g __attribute__((vector_size(16)));
typedef __attribute__((address_space(1))) v4i_g as1_v4i;
typedef __attribute__((address_space(3))) v4i_g as3_v4i;

__device__ __forceinline__ as1_v4i* sce_to_global(const void* p) {
    return (as1_v4i*)(unsigned long long)p;            // int -> AS1 ptr (64-bit)
}
__device__ __forceinline__ as3_v4i* sce_to_lds(const void* p) {
    // generic shared pointer: low 32 bits are the LDS byte offset
    return (as3_v4i*)(unsigned)(unsigned long long)p;  // int -> AS3 ptr (32-bit)
}
#endif

__global__ void sce_zero_kernel(float* __restrict__ gConf, unsigned* __restrict__ gAcc) {
    int i = blockIdx.x * blockDim.x + threadIdx.x;
    if (i < SCE_NSEG) { gConf[i] = 0.0f; gAcc[i] = 0u; }
}

__global__ void __launch_bounds__(BLOCK_THREADS)
sce_hist_kernel(const float* __restrict__ logits,
                const int*   __restrict__ labels,
                float* __restrict__ gConf,
                unsigned* __restrict__ gAcc,
                int B)
{
    __shared__ float    sConf[SCE_NSEG];
    __shared__ unsigned sAcc[SCE_NSEG];
#if SCE_ASYNC
    __shared__ float    stage[WAVES_PER_BLOCK][SCE_DEPTH][SCE_C];
#endif

    for (int i = threadIdx.x; i < SCE_NSEG; i += BLOCK_THREADS) {
        sConf[i] = 0.0f;
        sAcc[i]  = 0u;
    }
    __syncthreads();

    const int lane = threadIdx.x & 31;
    const int wv   = threadIdx.x >> 5;
    const int gw   = blockIdx.x * WAVES_PER_BLOCK + wv;   // global wave id
    const int nw   = gridDim.x * WAVES_PER_BLOCK;         // total waves

#if SCE_ASYNC
    // ---- prologue: fill the ring (up to DEPTH-1 loads in flight) ----
    #pragma unroll
    for (int d = 0; d < SCE_DEPTH - 1; ++d) {
        const int r = gw + d * nw;
        if (r < B) {
            __builtin_amdgcn_global_load_async_to_lds_b128(
                sce_to_global(logits + (size_t)r * SCE_C + lane * 4),
                sce_to_lds(&stage[wv][d][lane * 4]), 0, 0);
        }
    }
#endif

    int iter = 0;
    for (int row = gw; row < B; row += nw, ++iter) {
#if SCE_ASYNC
        const int nxtIss = row + (SCE_DEPTH - 1) * nw;
        if (nxtIss < B) {
            __builtin_amdgcn_global_load_async_to_lds_b128(
                sce_to_global(logits + (size_t)nxtIss * SCE_C + lane * 4),
                sce_to_lds(&stage[wv][(iter + SCE_DEPTH - 1) & (SCE_DEPTH - 1)][lane * 4]),
                0, 0);
            // in-order completion: allowing DEPTH-1 outstanding guarantees the
            // buffer consumed below has landed
            __builtin_amdgcn_s_wait_asynccnt(SCE_DEPTH - 1);
        } else {
            __builtin_amdgcn_s_wait_asynccnt(0);       // ragged tail: drain
        }
        v4f v = *(const v4f*)&stage[wv][iter & (SCE_DEPTH - 1)][lane * 4];
#else
        const int nxt = row + nw;
        if (nxt < B) {
            __builtin_prefetch(logits + (size_t)nxt * SCE_C + lane * 4, 0, 0);
        }
        v4f v = __builtin_nontemporal_load(
            (const v4f*)(logits + (size_t)row * SCE_C + lane * 4));
#endif

        const int lab = labels[row];

        // ---- wave32 softmax: max reduce ----
        float m = fmaxf(fmaxf(v.x, v.y), fmaxf(v.z, v.w));
        #pragma unroll
        for (int off = 16; off > 0; off >>= 1)
            m = fmaxf(m, __shfl_xor(m, off, 32));

        float e0 = __expf(v.x - m);
        float e1 = __expf(v.y - m);
        float e2 = __expf(v.z - m);
        float e3 = __expf(v.w - m);

        float s = (e0 + e1) + (e2 + e3);
        #pragma unroll
        for (int off = 16; off > 0; off >>= 1)
            s += __shfl_xor(s, off, 32);

        const float inv = 1.0f / s;
        float p[4] = { e0 * inv, e1 * inv, e2 * inv, e3 * inv };

        // ---- bin + LDS-privatized segment accumulation ----
        #pragma unroll
        for (int j = 0; j < 4; ++j) {
            const float pj = p[j];
            if (pj > 0.0f) {                         // matches reference "valid"
                const int c = lane * 4 + j;
                int b = (int)ceilf(pj * 15.0f) - 1;  // (b/15, (b+1)/15] bin
                b = b < 0 ? 0 : (b > SCE_NB - 1 ? SCE_NB - 1 : b);
                const int seg = c * SCE_NB + b;
                atomicAdd(&sConf[seg], pj);          // ds_add_f32
                if (c == lab)
                    atomicAdd(&sAcc[seg], 1u);       // ds_add_u32
            }
        }
    }

    __syncthreads();

    // ---- merge block-private histogram into global scratch ----
    for (int i = threadIdx.x; i < SCE_NSEG; i += BLOCK_THREADS) {
        const float cv = sConf[i];
        if (cv != 0.0f) atomicAdd(&gConf[i], cv);    // global_atomic_add_f32
        const unsigned av = sAcc[i];
        if (av)         atomicAdd(&gAcc[i], av);     // global_atomic_add_u32
    }
}

__global__ void __launch_bounds__(BLOCK_THREADS)
sce_final_kernel(const float* __restrict__ gConf,
                 const unsigned* __restrict__ gAcc,
                 float* __restrict__ out,
                 int B)
{
    __shared__ float red[WAVES_PER_BLOCK];

    // counts>0 gate is redundant: cnt==0 => conf==0 && acc==0 => |conf-acc|==0
    float t = 0.0f;
    for (int i = threadIdx.x; i < SCE_NSEG; i += BLOCK_THREADS)
        t += fabsf(gConf[i] - (float)gAcc[i]);

    #pragma unroll
    for (int off = 16; off > 0; off >>= 1)
        t += __shfl_xor(t, off, 32);

    if ((threadIdx.x & 31) == 0)
        red[threadIdx.x >> 5] = t;
    __syncthreads();

    if (threadIdx.x < WAVES_PER_BLOCK) {
        float r = red[threadIdx.x];
        #pragma unroll
        for (int off = WAVES_PER_BLOCK / 2; off > 0; off >>= 1)
            r += __shfl_xor(r, off, WAVES_PER_BLOCK);
        if (threadIdx.x == 0)
            out[0] = r / (float)B / (float)SCE_C;
    }
}

extern "C" void kernel_launch(void* const* d_in, const int* in_sizes, int n_in,
                              void* d_out, int out_size, void* d_ws, size_t ws_size,
                              hipStream_t stream) {
    const float* logits = (const float*)d_in[0];
    const int*   labels = (const int*)d_in[1];
    const int B = in_sizes[1];                 // labels count == number of rows

    float*    gConf = (float*)d_ws;
    unsigned* gAcc  = (unsigned*)((char*)d_ws + SCE_NSEG * sizeof(float));

    sce_zero_kernel<<<(SCE_NSEG + BLOCK_THREADS - 1) / BLOCK_THREADS,
                      BLOCK_THREADS, 0, stream>>>(gConf, gAcc);

    const int blocks = 1024;                   // 8192 waves, 64 rows/wave
    sce_hist_kernel<<<blocks, BLOCK_THREADS, 0, stream>>>(logits, labels,
                                                          gConf, gAcc, B);

    sce_final_kernel<<<1, BLOCK_THREADS, 0, stream>>>(gConf, gAcc,
                                                      (float*)d_out, B);
}